// GCNII_31104153158281
// MI455X (gfx1250) — compile-verified
//
#include <hip/hip_runtime.h>
#include <math.h>

#define NN      50000
#define NE      800000
#define NFEAT   512
#define NHID    128
#define NLAYERS 16
#define LAMDA   0.5f
#define ALPHA_R 0.1f

typedef float v2f __attribute__((ext_vector_type(2)));
typedef float v8f __attribute__((ext_vector_type(8)));

// ---------------------------------------------------------------------------
// gfx1250 async global->LDS helpers (ASYNCcnt-tracked DMA, no VGPR round-trip)
// ---------------------------------------------------------------------------
__device__ __forceinline__ void async_load_b128(unsigned lds_byte_off, const float* gaddr) {
    asm volatile("global_load_async_to_lds_b128 %0, %1, off"
                 :: "v"(lds_byte_off), "v"(gaddr) : "memory");
}
__device__ __forceinline__ void wait_asynccnt0() {
    asm volatile("s_wait_asynccnt 0" ::: "memory");
}
__device__ __forceinline__ unsigned lds_offset_of(const void* p) {
    // generic LDS address = {shared aperture, 32-bit LDS offset}; low 32 bits
    return (unsigned)(unsigned long long)p;
}

// ---------------------------------------------------------------------------
// Graph preprocessing
// ---------------------------------------------------------------------------
__global__ void count_kernel(const int* __restrict__ eidx, int* __restrict__ cnt, int ne) {
    int e = blockIdx.x * blockDim.x + threadIdx.x;
    if (e < ne) atomicAdd(&cnt[eidx[e]], 1);
}

#define SCAN_T 1024
#define ITEMS  49   // 1024*49 = 50176 >= 50000
__global__ void scan_kernel(const int* __restrict__ cnt, int* __restrict__ rowptr,
                            float* __restrict__ dinv, int n) {
    __shared__ int sdata[SCAN_T];
    int t = threadIdx.x;
    int base = t * ITEMS;
    int tot = 0;
    for (int i = 0; i < ITEMS; ++i) {
        int idx = base + i;
        if (idx < n) tot += cnt[idx];
    }
    sdata[t] = tot;
    __syncthreads();
    for (int off = 1; off < SCAN_T; off <<= 1) {
        int v = (t >= off) ? sdata[t - off] : 0;
        __syncthreads();
        sdata[t] += v;
        __syncthreads();
    }
    int run = sdata[t] - tot;  // exclusive prefix
    for (int i = 0; i < ITEMS; ++i) {
        int idx = base + i;
        if (idx < n) {
            rowptr[idx] = run;
            int c = cnt[idx];
            run += c;
            dinv[idx] = rsqrtf((float)c + 1.0f);  // deg includes self loop
        }
    }
    if (t == SCAN_T - 1) rowptr[n] = sdata[SCAN_T - 1];
}

__global__ void fill_kernel(const int* __restrict__ eidx, const int* __restrict__ rowptr,
                            int* __restrict__ fill, const float* __restrict__ dinv,
                            int* __restrict__ csr_col, float* __restrict__ csr_w, int ne) {
    int e = blockIdx.x * blockDim.x + threadIdx.x;
    if (e >= ne) return;
    int r = eidx[e];
    int c = eidx[ne + e];
    int pos = rowptr[r] + atomicAdd(&fill[r], 1);
    csr_col[pos] = c;
    csr_w[pos]   = dinv[r] * dinv[c];
}

// ---------------------------------------------------------------------------
// CSR SpMM + residual mix:  S[r] = 0.9*(A_hat h)[r] + 0.1*h0[r]
// One graph row handled by 128 threads (one per feature); fully coalesced.
// ---------------------------------------------------------------------------
__global__ void spmm_kernel(const float* __restrict__ h, const float* __restrict__ h0,
                            const int* __restrict__ rowptr, const int* __restrict__ csr_col,
                            const float* __restrict__ csr_w, const float* __restrict__ dinv,
                            float* __restrict__ S, int n) {
    int f    = threadIdx.x & 127;
    int rloc = threadIdx.x >> 7;        // 0..1
    int base = blockIdx.x * 16;
    for (int i = 0; i < 8; ++i) {
        int r = base + i * 2 + rloc;
        if (r >= n) continue;
        float di  = dinv[r];
        float acc = di * di * h[(size_t)r * NHID + f];   // self loop
        int s = rowptr[r], e = rowptr[r + 1];
        for (int j = s; j < e; ++j) {
            acc += csr_w[j] * h[(size_t)csr_col[j] * NHID + f];
        }
        S[(size_t)r * NHID + f] = (1.0f - ALPHA_R) * acc + ALPHA_R * h0[(size_t)r * NHID + f];
    }
}

// ---------------------------------------------------------------------------
// WMMA fp32 GEMM:  out = act( alpha * (A @ W) + beta * A + bias )
// N fixed = 128. Block: 256 threads = 8 waves, tile 64 rows x 128 cols.
// Wave w: row sub-tile (w>>1), col half (w&1)*64 (4 tiles of 16).
// A staged via async global->LDS DMA; W staged transposed so B fragments are
// contiguous b64 reads (conflict-free with stride 68 == 4 mod 64).
// ---------------------------------------------------------------------------
#define KC   64
#define ASTR 68    // LDS row stride for A (floats)
#define WSTR 68    // LDS row stride for W^T (floats)

__global__ void gemm_wmma(const float* __restrict__ A, const float* __restrict__ W,
                          const float* __restrict__ bias, float* __restrict__ out,
                          int M, int K, float alpha, float beta, int relu) {
    __shared__ float As[64 * ASTR];       // [row 0..63][k 0..KC-1]
    __shared__ float WsT[NHID * WSTR];    // [col 0..127][k 0..KC-1]

    const int tid  = threadIdx.x;
    const int lane = tid & 31;
    const int wave = tid >> 5;
    const int rt   = wave >> 1;          // row sub-tile 0..3
    const int cb   = (wave & 1) * 64;    // col base 0 or 64
    const int brow = blockIdx.x * 64;

    const int mloc = lane & 15;
    const int koff = (lane >> 4) << 1;   // 0 or 2

    v8f acc[4];
    #pragma unroll
    for (int c = 0; c < 4; ++c) acc[c] = (v8f)0.0f;

    for (int k0 = 0; k0 < K; k0 += KC) {
        // --- stage A chunk (64 x KC) via async DMA to LDS ---
        for (int i = tid; i < 64 * (KC / 4); i += 256) {
            int r  = i >> 4;
            int c4 = i & 15;
            int gr = brow + r;
            if (gr >= M) gr = M - 1;
            async_load_b128(lds_offset_of(&As[r * ASTR + c4 * 4]),
                            A + (size_t)gr * K + k0 + c4 * 4);
        }
        // --- stage W chunk transposed: WsT[n][k] = W[k0+k][n] ---
        for (int i = tid; i < KC * NHID; i += 256) {
            int r = i >> 7;       // k 0..KC-1
            int n = i & 127;      // col
            WsT[n * WSTR + r] = W[(size_t)(k0 + r) * NHID + n];
        }
        wait_asynccnt0();
        __syncthreads();

        #pragma unroll
        for (int kk = 0; kk < KC; kk += 4) {
            // A fragment: lanes 0-15 -> K=kk+0/1, lanes 16-31 -> K=kk+2/3
            v2f a = *(const v2f*)(&As[(rt * 16 + mloc) * ASTR + kk + koff]);
            #pragma unroll
            for (int c = 0; c < 4; ++c) {
                int nidx = cb + c * 16 + mloc;
                v2f b = *(const v2f*)(&WsT[nidx * WSTR + kk + koff]);
                acc[c] = __builtin_amdgcn_wmma_f32_16x16x4_f32(
                    false, a, false, b, (short)0, acc[c], false, false);
            }
        }
        __syncthreads();
    }

    // --- epilogue ---
    const int mhi = (lane >> 4) << 3;    // 0 or 8
    #pragma unroll
    for (int c = 0; c < 4; ++c) {
        int n  = cb + c * 16 + mloc;
        float bv = bias ? bias[n] : 0.0f;
        #pragma unroll
        for (int v = 0; v < 8; ++v) {
            int gr = brow + rt * 16 + v + mhi;
            if (gr < M) {
                float val = alpha * acc[c][v] + bv;
                if (beta != 0.0f) val += beta * A[(size_t)gr * K + n];  // K==128 here
                if (relu) val = fmaxf(val, 0.0f);
                out[(size_t)gr * NHID + n] = val;
            }
        }
    }
}

// ---------------------------------------------------------------------------
// Host launcher
// ---------------------------------------------------------------------------
extern "C" void kernel_launch(void* const* d_in, const int* in_sizes, int n_in,
                              void* d_out, int out_size, void* d_ws, size_t ws_size,
                              hipStream_t stream) {
    const float* x      = (const float*)d_in[0];
    const int*   eidx   = (const int*)d_in[1];
    const float* conv_w = (const float*)d_in[2];
    const float* fc0_w  = (const float*)d_in[3];
    const float* fc0_b  = (const float*)d_in[4];
    const float* fc1_w  = (const float*)d_in[5];
    const float* fc1_b  = (const float*)d_in[6];
    float* outp = (float*)d_out;

    char* ws = (char*)d_ws;
    size_t off = 0;
    auto alloc = [&](size_t bytes) -> char* {
        char* p = ws + off;
        off = (off + bytes + 255) & ~(size_t)255;
        return p;
    };
    int*   cnt     = (int*)  alloc((size_t)NN * 4);
    int*   fill    = (int*)  alloc((size_t)NN * 4);
    int*   rowptr  = (int*)  alloc((size_t)(NN + 1) * 4);
    float* dinv    = (float*)alloc((size_t)NN * 4);
    int*   csr_col = (int*)  alloc((size_t)NE * 4);
    float* csr_w   = (float*)alloc((size_t)NE * 4);
    float* h0      = (float*)alloc((size_t)NN * NHID * 4);
    float* hA      = (float*)alloc((size_t)NN * NHID * 4);
    float* hB      = (float*)alloc((size_t)NN * NHID * 4);
    float* Sb      = (float*)alloc((size_t)NN * NHID * 4);

    hipMemsetAsync(cnt,  0, (size_t)NN * 4, stream);
    hipMemsetAsync(fill, 0, (size_t)NN * 4, stream);

    const int EB = (NE + 255) / 256;           // 3125
    count_kernel<<<EB, 256, 0, stream>>>(eidx, cnt, NE);
    scan_kernel<<<1, SCAN_T, 0, stream>>>(cnt, rowptr, dinv, NN);
    fill_kernel<<<EB, 256, 0, stream>>>(eidx, rowptr, fill, dinv, csr_col, csr_w, NE);

    const int GB = (NN + 63) / 64;             // 782 blocks for GEMM
    const int SB = (NN + 15) / 16;             // 3125 blocks for SpMM

    // h0 = relu(x @ fc0_w + fc0_b)
    gemm_wmma<<<GB, 256, 0, stream>>>(x, fc0_w, fc0_b, h0, NN, NFEAT, 1.0f, 0.0f, 1);

    const float* hcur = h0;
    for (int l = 0; l < NLAYERS; ++l) {
        spmm_kernel<<<SB, 256, 0, stream>>>(hcur, h0, rowptr, csr_col, csr_w, dinv, Sb, NN);
        float theta = logf(LAMDA / (float)(l + 1) + 1.0f);
        float* hn = (l & 1) ? hB : hA;
        gemm_wmma<<<GB, 256, 0, stream>>>(Sb, conv_w + (size_t)l * NHID * NHID, nullptr,
                                          hn, NN, NHID, theta, 1.0f - theta, 1);
        hcur = hn;
    }

    // out = h @ fc1_w + fc1_b
    gemm_wmma<<<GB, 256, 0, stream>>>(hcur, fc1_w, fc1_b, outp, NN, NHID, 1.0f, 0.0f, 0);
}